// YoloV3Loss_83296595738880
// MI455X (gfx1250) — compile-verified
//
#include <hip/hip_runtime.h>
#include <math.h>

#define A_NUM 3
#define NCLS 80
#define NCH  (5 + NCLS)      // 85
#define BATCH 32
#define NGT 20
#define THRS 0.5f
#define LOG_MIN -100.0f

typedef __attribute__((ext_vector_type(2))) float v2f;
typedef __attribute__((ext_vector_type(8))) float v8f;

// Exact f32 sum of all 32 lanes of the wave using V_WMMA_F32_16X16X4_F32.
// A[m][k]: lane m (0-15) holds K=0 in vgpr0, lanes 16-31 hold K=2 in vgpr0
// (vgpr1 = K=1/K=3 set to zero). B = all ones -> D[m][n] = v(m) + v(m+16).
// Summing the 8 D vgprs gives sum over M=0..7 (lanes 0-15) / M=8..15
// (lanes 16-31); one shfl_xor(16) combines the halves. All lanes return the
// full 32-lane sum. Requires EXEC == all ones (callers guarantee this).
__device__ __forceinline__ float wave_reduce_wmma(float v) {
  v2f a; a[0] = v;    a[1] = 0.0f;
  v2f b; b[0] = 1.0f; b[1] = 1.0f;
  v8f c = {};
  c = __builtin_amdgcn_wmma_f32_16x16x4_f32(false, a, false, b,
                                            (short)0, c, false, false);
  float s = ((c[0] + c[1]) + (c[2] + c[3])) + ((c[4] + c[5]) + (c[6] + c[7]));
  s += __shfl_xor(s, 16, 32);
  return s;
}

__device__ __forceinline__ float sigmoidf(float z) {
  return 1.0f / (1.0f + expf(-z));
}

// One thread per anchor slot: slot = ((b*G + row)*G + col)*A + a.
// Fuses: noobj test (20 IoUs vs this slot's anchor), GT->cell assignment
// (recomputed per slot; last matching GT wins), and the per-slot loss terms.
__global__ void yolo_scale_loss_kernel(
    const float* __restrict__ pred,     // (B, G, G, A*85)
    const float* __restrict__ gtb,      // (B, 20, 5) xyxy + cls
    const float* __restrict__ anchors,  // (G, G, A, 4) cx,cy,w,h (grid units)
    float* __restrict__ partials, int part_off,
    int G, float invGG)
{
  const int slot  = blockIdx.x * blockDim.x + threadIdx.x;
  const int total = BATCH * G * G * A_NUM;
  float partial = 0.0f;

  if (slot < total) {
    const int a    = slot % A_NUM;
    const int cell = slot / A_NUM;
    const int col  = cell % G;
    const int rc   = cell / G;
    const int row  = rc % G;
    const int b    = rc / G;

    // The 3 anchor (w,h) pairs of this cell.
    const float* abase = anchors + (size_t)((row * G + col) * A_NUM) * 4;
    const float aw0 = abase[0 * 4 + 2], ah0 = abase[0 * 4 + 3];
    const float aw1 = abase[1 * 4 + 2], ah1 = abase[1 * 4 + 3];
    const float aw2 = abase[2 * 4 + 2], ah2 = abase[2 * 4 + 3];

    const float my_aw = (a == 0) ? aw0 : ((a == 1) ? aw1 : aw2);
    const float my_ah = (a == 0) ? ah0 : ((a == 1) ? ah1 : ah2);
    const float acx = (float)col + 0.5f;
    const float acy = (float)row + 0.5f;
    const float ax0 = acx - 0.5f * my_aw, ay0 = acy - 0.5f * my_ah;
    const float ax1 = acx + 0.5f * my_aw, ay1 = acy + 0.5f * my_ah;
    const float areaA = my_aw * my_ah;

    bool  hit = false;   // any GT IoU >= THR against this anchor -> not noobj
    bool  obj = false;
    float tx = 0.0f, ty = 0.0f, tw = 0.0f, th = 0.0f, bscale = 1.0f;
    int   tcls = 1;

    const float fG = (float)G;
    for (int n = 0; n < NGT; ++n) {
      const float* g = gtb + (size_t)(b * NGT + n) * 5;
      const float X0 = g[0] * fG, Y0 = g[1] * fG;
      const float X2 = g[2] * fG, Y2 = g[3] * fG;
      const float areaG = (X2 - X0) * (Y2 - Y0);

      // IoU of this GT with this slot's anchor (for noobj mask).
      float iw = fminf(X2, ax1) - fmaxf(X0, ax0); iw = fmaxf(iw, 0.0f);
      float ih = fminf(Y2, ay1) - fmaxf(Y0, ay0); ih = fmaxf(ih, 0.0f);
      const float inter = iw * ih;
      const float iou = inter / (areaG + areaA - inter);
      if (iou >= THRS) hit = true;

      // Cell assignment: does this GT's center land in my cell?
      const float cx = 0.5f * (X0 + X2);
      const float cy = 0.5f * (Y0 + Y2);
      const int gcol = (int)floorf(cx);
      const int grow = (int)floorf(cy);
      if (gcol == col && grow == row) {
        float best = -1.0f; int ba = 0; float baw = aw0, bah = ah0;
#define CELL_IOU(AW, AH, IDX)                                               \
        {                                                                   \
          const float hx0 = acx - 0.5f * (AW), hy0 = acy - 0.5f * (AH);     \
          const float hx1 = acx + 0.5f * (AW), hy1 = acy + 0.5f * (AH);     \
          float w_ = fminf(X2, hx1) - fmaxf(X0, hx0); w_ = fmaxf(w_, 0.f);  \
          float h_ = fminf(Y2, hy1) - fmaxf(Y0, hy0); h_ = fmaxf(h_, 0.f);  \
          const float in_ = w_ * h_;                                        \
          const float io_ = in_ / (areaG + (AW) * (AH) - in_);              \
          if (io_ > best) { best = io_; ba = IDX; baw = (AW); bah = (AH); } \
        }
        CELL_IOU(aw0, ah0, 0)
        CELL_IOU(aw1, ah1, 1)
        CELL_IOU(aw2, ah2, 2)
#undef CELL_IOU
        if (best >= THRS && ba == a) {    // last matching GT wins (serial n)
          obj = true;
          tx = cx - (float)gcol;
          ty = cy - (float)grow;
          const float cw = X2 - X0, ch = Y2 - Y0;
          tw = logf(cw / baw + 1e-16f);
          th = logf(ch / bah + 1e-16f);
          tcls = (int)(g[4] - 1.0f);
          bscale = 2.0f - cw * ch * invGG;
        }
      }
    }

    // Per-slot loss terms from the 85 prediction channels.
    const float* pp = pred + (size_t)((b * G + row) * G + col) * (A_NUM * NCH)
                           + (size_t)a * NCH;
    const float x    = sigmoidf(pp[0]);
    const float y    = sigmoidf(pp[1]);
    const float w    = tanhf(pp[2]);
    const float h    = tanhf(pp[3]);
    const float conf = sigmoidf(pp[4]);

    // log_softmax over 80 classes: pass 1 max (+ capture target logit),
    // pass 2 sum-exp (second pass hits WGP$).
    float m = -INFINITY, lt = 0.0f;
    for (int cc = 0; cc < NCLS; ++cc) {
      const float l = pp[5 + cc];
      m = fmaxf(m, l);
      if (cc == tcls) lt = l;
    }
    float se = 0.0f;
    for (int cc = 0; cc < NCLS; ++cc) se += expf(pp[5 + cc] - m);
    const float lse = m + logf(se);

    const float objf   = obj ? 1.0f : 0.0f;
    const float noobjf = hit ? 0.0f : 1.0f;

    const float dx = x - tx, dy = y - ty, dw = w - tw, dh = h - th;
    const float lbox = objf * bscale * (dx * dx + dy * dy + dw * dw + dh * dh);
    const float lcls = -objf * (lt - lse);
    const float log_c   = fmaxf(logf(conf),     LOG_MIN);
    const float log_1mc = fmaxf(log1pf(-conf),  LOG_MIN);
    const float bce = -(objf * log_c + (1.0f - objf) * log_1mc);
    const float lconf = (objf + noobjf) * bce;

    partial = lbox + lcls + lconf;
  }

  // Block reduction: WMMA per-wave (EXEC reconverged, all-ones), then 8->1.
  const float wsum = wave_reduce_wmma(partial);
  __shared__ float smem[8];
  const int lane = threadIdx.x & 31;
  const int wid  = threadIdx.x >> 5;
  if (lane == 0) smem[wid] = wsum;
  __syncthreads();
  if (threadIdx.x == 0) {
    float t = 0.0f;
    for (int k = 0; k < 8; ++k) t += smem[k];
    partials[part_off + blockIdx.x] = t;
  }
}

__global__ void yolo_final_reduce_kernel(const float* __restrict__ partials,
                                         int n, float* __restrict__ out)
{
  float v = 0.0f;
  for (int i = threadIdx.x; i < n; i += blockDim.x) v += partials[i];
  const float wsum = wave_reduce_wmma(v);   // every wave, EXEC all ones
  __shared__ float smem[8];
  const int lane = threadIdx.x & 31;
  const int wid  = threadIdx.x >> 5;
  if (lane == 0) smem[wid] = wsum;
  __syncthreads();
  if (threadIdx.x == 0) {
    float t = 0.0f;
    for (int k = 0; k < 8; ++k) t += smem[k];
    out[0] = t;
  }
}

extern "C" void kernel_launch(void* const* d_in, const int* in_sizes, int n_in,
                              void* d_out, int out_size, void* d_ws, size_t ws_size,
                              hipStream_t stream) {
  (void)in_sizes; (void)n_in; (void)out_size; (void)ws_size;

  const float* pred0 = (const float*)d_in[0];
  const float* pred1 = (const float*)d_in[1];
  const float* pred2 = (const float*)d_in[2];
  const float* gtb   = (const float*)d_in[3];
  const float* anch0 = (const float*)d_in[4];
  const float* anch1 = (const float*)d_in[5];
  const float* anch2 = (const float*)d_in[6];
  float* out = (float*)d_out;
  float* partials = (float*)d_ws;     // (nb0+nb1+nb2) floats, ~5.3 KB

  const int G0 = 13, G1 = 26, G2 = 52;
  const int t0 = BATCH * G0 * G0 * A_NUM;   // 16224
  const int t1 = BATCH * G1 * G1 * A_NUM;   // 64896
  const int t2 = BATCH * G2 * G2 * A_NUM;   // 259584
  const int nb0 = (t0 + 255) / 256;
  const int nb1 = (t1 + 255) / 256;
  const int nb2 = (t2 + 255) / 256;

  yolo_scale_loss_kernel<<<nb0, 256, 0, stream>>>(
      pred0, gtb, anch0, partials, 0, G0, 1.0f / (float)(G0 * G0));
  yolo_scale_loss_kernel<<<nb1, 256, 0, stream>>>(
      pred1, gtb, anch1, partials, nb0, G1, 1.0f / (float)(G1 * G1));
  yolo_scale_loss_kernel<<<nb2, 256, 0, stream>>>(
      pred2, gtb, anch2, partials, nb0 + nb1, G2, 1.0f / (float)(G2 * G2));

  yolo_final_reduce_kernel<<<1, 256, 0, stream>>>(
      partials, nb0 + nb1 + nb2, out);
}